// DQN_67611375174034
// MI455X (gfx1250) — compile-verified
//
#include <hip/hip_runtime.h>
#include <hip/hip_bf16.h>

typedef float v2f __attribute__((ext_vector_type(2)));
typedef float v4f __attribute__((ext_vector_type(4)));
typedef float v8f __attribute__((ext_vector_type(8)));

#define N_NODES   8192
#define N_EDGES   262144
#define HID       128
#define FLATN     (N_NODES * HID)       // 1048576
#define FC1_PARTS 2048
#define ROWS_PER_PART (FLATN / FC1_PARTS)  // 512

// LDS byte offset = low 32 bits of the generic pointer (ISA 10.2: LDS aperture
// truncates generic address to addr[31:0]).
__device__ static inline unsigned lds_byte_offset(const void* p) {
    return (unsigned)(unsigned long long)p;
}

// ---------------------------------------------------------------------------
// Dense GEMM: out[N_NODES x 128] = A[N_NODES x 128] * W[128 x 128]
// One block = 128 threads = 4 waves; each wave owns a 16-row strip.
// W staged in LDS (64 KB) via ASYNC global->LDS DMA (no VGPR round-trip).
// fp32 WMMA: V_WMMA_F32_16X16X4_F32, K=128 -> 32 steps per 16x16 tile.
// ---------------------------------------------------------------------------
__global__ __launch_bounds__(128)
void gemm_nodes_wmma(const float* __restrict__ A, const float* __restrict__ W,
                     float* __restrict__ out) {
    __shared__ float Wlds[HID * HID];

    const int tid = threadIdx.x;
    // Async DMA of W into LDS: 4096 x b128 transfers, 32 per thread.
    // Tracked with ASYNCcnt; each wave issues 32 (<= 63 counter capacity).
    {
        #pragma unroll 1
        for (int i = 0; i < 32; ++i) {
            const int idx = i * 128 + tid;             // float4 index
            unsigned     lds = lds_byte_offset(&Wlds[idx * 4]);
            const float* g   = W + (size_t)idx * 4;
            asm volatile("global_load_async_to_lds_b128 %0, %1, off"
                         :: "v"(lds), "v"(g) : "memory");
        }
        asm volatile("s_wait_asynccnt 0x0" ::: "memory");
    }
    __syncthreads();

    const int wave = tid >> 5;
    const int lane = tid & 31;
    const int half = lane >> 4;   // selects K-pair for A/B frags, M-half for C
    const int l16  = lane & 15;   // M row for A frag; N col for B/C frags
    const int rowBase = blockIdx.x * 64 + wave * 16;

    // Hoist all A fragments (16x4 f32 tile = v2f per lane) for K = 0..127.
    v2f afrag[32];
    #pragma unroll
    for (int ks = 0; ks < 32; ++ks) {
        const float* ap = A + (size_t)(rowBase + l16) * HID + ks * 4 + 2 * half;
        afrag[ks] = *(const v2f*)ap;
    }

    #pragma unroll 1
    for (int nt = 0; nt < 8; ++nt) {
        const int ncol = nt * 16 + l16;
        v8f acc = {};
        #pragma unroll
        for (int ks = 0; ks < 32; ++ks) {
            const int k = ks * 4 + 2 * half;
            v2f b;
            b.x = Wlds[k * HID + ncol];
            b.y = Wlds[(k + 1) * HID + ncol];
            acc = __builtin_amdgcn_wmma_f32_16x16x4_f32(
                /*neg_a=*/false, afrag[ks], /*neg_b=*/false, b,
                /*c_mod=*/(short)0, acc, /*reuse_a=*/false, /*reuse_b=*/false);
        }
        #pragma unroll
        for (int v = 0; v < 8; ++v) {
            out[(size_t)(rowBase + half * 8 + v) * HID + ncol] = acc[v];
        }
    }
}

// ---------------------------------------------------------------------------
// Degree computation (self-loops => init to 1.0; +1 per incoming edge), then
// in-place rsqrt. Float adds of 1.0 are exact -> order independent.
// ---------------------------------------------------------------------------
__global__ void deg_init(float* __restrict__ deg) {
    int i = blockIdx.x * blockDim.x + threadIdx.x;
    if (i < N_NODES) deg[i] = 1.0f;
}

__global__ void deg_count(const int* __restrict__ ei, float* __restrict__ deg) {
    int e = blockIdx.x * blockDim.x + threadIdx.x;
    if (e < N_EDGES) atomicAdd(&deg[ei[N_EDGES + e]], 1.0f);
}

__global__ void deg_rsqrt(float* __restrict__ deg) {
    int i = blockIdx.x * blockDim.x + threadIdx.x;
    if (i < N_NODES) deg[i] = rsqrtf(fmaxf(deg[i], 1.0f));
}

// ---------------------------------------------------------------------------
// e[i,:] = dinv[i]^2 * h[i,:]   (self-loop term; also initializes buffer)
// ---------------------------------------------------------------------------
__global__ void self_init(const float* __restrict__ h, const float* __restrict__ dinv,
                          float* __restrict__ e) {
    int f = blockIdx.x * blockDim.x + threadIdx.x;   // over FLATN/4
    int node = f >> 5;
    float d = dinv[node];
    float s = d * d;
    v4f hv = ((const v4f*)h)[f];
    ((v4f*)e)[f] = hv * s;
}

// ---------------------------------------------------------------------------
// e[col,:] += dinv[row]*dinv[col] * h[row,:]   (32 threads/edge, 4 cols/thread)
// ---------------------------------------------------------------------------
__global__ void scatter_edges(const int* __restrict__ ei, const float* __restrict__ dinv,
                              const float* __restrict__ h, float* __restrict__ e) {
    int idx  = blockIdx.x * blockDim.x + threadIdx.x;  // N_EDGES * 32 threads
    int edge = idx >> 5;
    int c0   = (idx & 31) * 4;
    int r = ei[edge];
    int c = ei[N_EDGES + edge];
    float norm = dinv[r] * dinv[c];
    const v4f hv = *(const v4f*)(h + (size_t)r * HID + c0);
    float* dst = e + (size_t)c * HID + c0;
    atomicAdd(dst + 0, norm * hv.x);
    atomicAdd(dst + 1, norm * hv.y);
    atomicAdd(dst + 2, norm * hv.z);
    atomicAdd(dst + 3, norm * hv.w);
}

// ---------------------------------------------------------------------------
// e += bias (broadcast over nodes), optional ReLU
// ---------------------------------------------------------------------------
__global__ void bias_act(float* __restrict__ e, const float* __restrict__ b, int do_relu) {
    int f  = blockIdx.x * blockDim.x + threadIdx.x;  // over FLATN/4
    int c4 = (f & 31) * 4;
    v4f v = ((v4f*)e)[f];
    v += *(const v4f*)(b + c4);
    if (do_relu) {
        v.x = fmaxf(v.x, 0.0f); v.y = fmaxf(v.y, 0.0f);
        v.z = fmaxf(v.z, 0.0f); v.w = fmaxf(v.w, 0.0f);
    }
    ((v4f*)e)[f] = v;
}

// ---------------------------------------------------------------------------
// fc1 partial: partial[blk, :] = sum_{i in blk chunk} flat[i] * fc1_w[i, :]
// fc1_w is 536 MB (> 192 MB L2) -> stream with non-temporal b128 loads.
// 128 threads = 32 column-quads x 4 row-interleaved waves; each wave covers a
// full 512 B row per b128 instruction (fully coalesced); flat[i] is
// wave-uniform. Row-groups combined via LDS -> one partial row per block
// (fixed-order final reduce stays deterministic).
// ---------------------------------------------------------------------------
__global__ __launch_bounds__(128)
void fc1_partial(const float* __restrict__ flat, const float* __restrict__ w,
                 float* __restrict__ partial) {
    __shared__ v4f sred[128];
    const int tid = threadIdx.x;
    const int q   = tid & 31;          // column quad: cols 4q..4q+3
    const int rl  = tid >> 5;          // row interleave (= wave id)
    const int r0  = blockIdx.x * ROWS_PER_PART;

    v4f acc = {0.0f, 0.0f, 0.0f, 0.0f};
    #pragma unroll 4
    for (int r = rl; r < ROWS_PER_PART; r += 4) {
        const size_t i = (size_t)(r0 + r);
        __builtin_prefetch(w + (i + 32) * HID + q * 4, 0, 0);  // global_prefetch_b8
        v4f wv = __builtin_nontemporal_load((const v4f*)(w + i * HID) + q);
        acc += wv * flat[i];
    }
    sred[rl * 32 + q] = acc;
    __syncthreads();
    if (rl == 0) {
        v4f s = sred[q] + sred[32 + q] + sred[64 + q] + sred[96 + q];
        *((v4f*)(partial + (size_t)blockIdx.x * HID) + q) = s;
    }
}

// Fixed-order reduction of partials (deterministic), + bias, ReLU.
__global__ __launch_bounds__(128)
void fc1_reduce(const float* __restrict__ partial, const float* __restrict__ b,
                float* __restrict__ hvec) {
    const int j = threadIdx.x;
    float s = b[j];
    for (int p = 0; p < FC1_PARTS; ++p) s += partial[(size_t)p * HID + j];
    hvec[j] = fmaxf(s, 0.0f);
}

// ---------------------------------------------------------------------------
// Head: h2 = relu(h @ fc2_w + fc2_b); q = h2 @ fc3_w + fc3_b  (single block)
// ---------------------------------------------------------------------------
__global__ __launch_bounds__(128)
void head_kernel(const float* __restrict__ hvec,
                 const float* __restrict__ w2, const float* __restrict__ b2,
                 const float* __restrict__ w3, const float* __restrict__ b3,
                 float* __restrict__ out) {
    __shared__ float hs[HID];
    __shared__ float red[HID];
    const int j = threadIdx.x;
    hs[j] = hvec[j];
    __syncthreads();
    float s = b2[j];
    #pragma unroll 8
    for (int k = 0; k < HID; ++k) s = fmaf(hs[k], w2[k * HID + j], s);
    s = fmaxf(s, 0.0f);
    red[j] = s * w3[j];
    __syncthreads();
    #pragma unroll
    for (int off = 64; off > 0; off >>= 1) {
        if (j < off) red[j] += red[j + off];
        __syncthreads();
    }
    if (j == 0) out[0] = red[0] + b3[0];
}

// ---------------------------------------------------------------------------
extern "C" void kernel_launch(void* const* d_in, const int* in_sizes, int n_in,
                              void* d_out, int out_size, void* d_ws, size_t ws_size,
                              hipStream_t stream) {
    (void)in_sizes; (void)n_in; (void)out_size; (void)ws_size;

    const float* x     = (const float*)d_in[0];
    const int*   ei    = (const int*)  d_in[1];
    const float* W1    = (const float*)d_in[2];
    const float* b1    = (const float*)d_in[3];
    const float* W2    = (const float*)d_in[4];
    const float* b2    = (const float*)d_in[5];
    const float* fc1_w = (const float*)d_in[6];
    const float* fc1_b = (const float*)d_in[7];
    const float* fc2_w = (const float*)d_in[8];
    const float* fc2_b = (const float*)d_in[9];
    const float* fc3_w = (const float*)d_in[10];
    const float* fc3_b = (const float*)d_in[11];

    char* ws = (char*)d_ws;
    float* bufH    = (float*)(ws);                                  // 4 MB
    float* bufE    = (float*)(ws + (size_t)FLATN * 4);              // 4 MB
    float* dinv    = (float*)(ws + (size_t)FLATN * 8);              // 32 KB
    float* partial = (float*)(ws + (size_t)FLATN * 8 + 65536);      // 1 MB
    float* hvec    = (float*)(ws + (size_t)FLATN * 8 + 65536 +
                              (size_t)FC1_PARTS * HID * 4);         // 512 B

    // Degrees (with self-loop) -> dinv
    deg_init <<<N_NODES / 256, 256, 0, stream>>>(dinv);
    deg_count<<<N_EDGES / 256, 256, 0, stream>>>(ei, dinv);
    deg_rsqrt<<<N_NODES / 256, 256, 0, stream>>>(dinv);

    // Layer 1: h1 = x @ W1 ; e1 = scatter(norm * h1) + b1 ; ReLU
    gemm_nodes_wmma<<<N_NODES / 64, 128, 0, stream>>>(x, W1, bufH);
    self_init     <<<FLATN / 4 / 256, 256, 0, stream>>>(bufH, dinv, bufE);
    scatter_edges <<<N_EDGES * 32 / 256, 256, 0, stream>>>(ei, dinv, bufH, bufE);
    bias_act      <<<FLATN / 4 / 256, 256, 0, stream>>>(bufE, b1, 1);

    // Layer 2: h2 = e1 @ W2 ; e2 = scatter(norm * h2) + b2
    gemm_nodes_wmma<<<N_NODES / 64, 128, 0, stream>>>(bufE, W2, bufH);
    self_init     <<<FLATN / 4 / 256, 256, 0, stream>>>(bufH, dinv, bufE);
    scatter_edges <<<N_EDGES * 32 / 256, 256, 0, stream>>>(ei, dinv, bufH, bufE);
    bias_act      <<<FLATN / 4 / 256, 256, 0, stream>>>(bufE, b2, 0);

    // MLP head: fc1 (536 MB streamed, HBM-bound) -> fc2 -> fc3 -> scalar
    fc1_partial<<<FC1_PARTS, 128, 0, stream>>>(bufE, fc1_w, partial);
    fc1_reduce <<<1, 128, 0, stream>>>(partial, fc1_b, hvec);
    head_kernel<<<1, 128, 0, stream>>>(hvec, fc2_w, fc2_b, fc3_w, fc3_b, (float*)d_out);
}